// CapsuleLayer_15693810500146
// MI455X (gfx1250) — compile-verified
//
#include <hip/hip_runtime.h>
#include <hip/hip_bf16.h>

// CapsuleLayer dynamic routing, fused single kernel for MI455X (gfx1250).
// priors computed on the fly with V_WMMA_F32_16X16X4_F32 (exact f32),
// routing state (logits/probs/softmax/squash) kept entirely in LDS.
// x (9.4MB) + W (5.9MB) stay L2-resident across the 5 WMMA passes, so
// HBM traffic is ~15MB total and the kernel is L2/LDS bound, not HBM bound.

typedef __attribute__((ext_vector_type(2))) float v2f;
typedef __attribute__((ext_vector_type(8))) float v8f;

#define NCAP   10
#define NNODE  1152
#define DIN    8
#define DOUT   16
#define NBATCH 256
#define BTILE  16
#define NWAVES 8
#define KITERS (NNODE / NWAVES)  // 144, compile-time constant trip count

__device__ __forceinline__ v8f wmma4(v2f a, v2f b, v8f c) {
  // D = A(16x4 f32) * B(4x16 f32) + C(16x16 f32)
  return __builtin_amdgcn_wmma_f32_16x16x4_f32(false, a, false, b, (short)0, c,
                                               false, false);
}

// load the per-node A (x) and B (W) WMMA operands for node n
__device__ __forceinline__ void load_ab(const float* __restrict__ xbase,
                                        const float* __restrict__ wbase, int n,
                                        v2f& a0, v2f& a1, v2f& bm0, v2f& bm1) {
  const float* xp = xbase + n * DIN;
  const float* wp = wbase + n * (DIN * DOUT);
  __builtin_prefetch(wp + NWAVES * DIN * DOUT, 0, 0);
  a0 = *(const v2f*)xp;
  a1 = *(const v2f*)(xp + 4);
  bm0.x = wp[0];
  bm0.y = wp[DOUT];
  bm1.x = wp[4 * DOUT];
  bm1.y = wp[5 * DOUT];
}

// sum across the 16 lanes of each wave32 half (o-dimension reduction)
__device__ __forceinline__ float half16_sum(float t) {
  t += __shfl_xor(t, 1, 32);
  t += __shfl_xor(t, 2, 32);
  t += __shfl_xor(t, 4, 32);
  t += __shfl_xor(t, 8, 32);
  return t;
}

// softmax over the n (routing-node) dimension, per batch b; all in LDS
__device__ __forceinline__ void softmax_n(float* lg, float* probs, float* red,
                                          float* maxb, float* invZ, int tid) {
  const int b = tid & 15;
  const int seg = tid >> 4;
  float m = -3.4e38f;
  for (int n = seg; n < NNODE; n += 16) m = fmaxf(m, lg[n * 16 + b]);
  red[seg * 16 + b] = m;
  __syncthreads();
  if (tid < 16) {
    float mm = -3.4e38f;
#pragma unroll
    for (int s2 = 0; s2 < 16; ++s2) mm = fmaxf(mm, red[s2 * 16 + tid]);
    maxb[tid] = mm;
  }
  __syncthreads();
  const float mb = maxb[b];
  float z = 0.0f;
  for (int n = seg; n < NNODE; n += 16) {
    float e = __expf(lg[n * 16 + b] - mb);
    probs[n * 16 + b] = e;
    z += e;
  }
  red[seg * 16 + b] = z;
  __syncthreads();
  if (tid < 16) {
    float zz = 0.0f;
#pragma unroll
    for (int s2 = 0; s2 < 16; ++s2) zz += red[s2 * 16 + tid];
    invZ[tid] = 1.0f / zz;
  }
  __syncthreads();
  const float iz = invZ[b];
  for (int n = seg; n < NNODE; n += 16) probs[n * 16 + b] *= iz;
  __syncthreads();
}

// combine 8 wave accumulators -> s[b][o], then squash -> outArr[b][o]
__device__ __forceinline__ void reduce_squash(const v8f acc, float scale,
                                              float* sred, float* sArr,
                                              float* outArr, float* scaleArr,
                                              int tid, int lane, int wave) {
#pragma unroll
  for (int j = 0; j < 8; ++j) sred[j * 256 + wave * 32 + lane] = acc[j];
  __syncthreads();
  {
    const int j_ = tid >> 5;
    const int lane_ = tid & 31;
    float s = 0.0f;
#pragma unroll
    for (int w = 0; w < NWAVES; ++w) s += sred[j_ * 256 + w * 32 + lane_];
    s *= scale;
    sArr[(j_ + 8 * (lane_ >> 4)) * 16 + (lane_ & 15)] = s;
  }
  __syncthreads();
  if (tid < 16) {
    float ss = 0.0f;
#pragma unroll
    for (int o = 0; o < 16; ++o) {
      float v = sArr[tid * 16 + o];
      ss += v * v;
    }
    // squash scale = ||s||^2/(1+||s||^2) / ||s||
    scaleArr[tid] = (ss / (1.0f + ss)) * rsqrtf(ss);
  }
  __syncthreads();
  outArr[tid] = sArr[tid] * scaleArr[tid >> 4];
  __syncthreads();
}

__global__ __launch_bounds__(256) void capsule_route_kernel(
    const float* __restrict__ x, const float* __restrict__ W,
    float* __restrict__ out) {
  __shared__ float lg[NNODE * 16];     // logits [n][b]
  __shared__ float probs[NNODE * 16];  // softmax probs [n][b]
  __shared__ float sred[NWAVES * 256];
  __shared__ float sArr[256];
  __shared__ float outArr[256];
  __shared__ float red[256];
  __shared__ float maxb[16];
  __shared__ float invZ[16];
  __shared__ float scaleArr[16];

  const int tid = threadIdx.x;
  const int lane = tid & 31;
  const int wave = tid >> 5;
  const int laneLo = lane & 15;  // o column / batch row in A
  const int half = lane >> 4;    // K half for A/B operands
  const int c = blockIdx.x >> 4;
  const int b0 = (blockIdx.x & 15) * BTILE;

  // A: lane holds x[b0+laneLo, n, half*2 + {0,1}] (and +4 for 2nd WMMA)
  const float* xbase = x + (size_t)(b0 + laneLo) * NNODE * DIN + half * 2;
  // B: lane holds W[c, n, half*2 + {0,1}, laneLo] (and +4 for 2nd WMMA)
  const float* wbase =
      W + (size_t)c * NNODE * (DIN * DOUT) + half * 2 * DOUT + laneLo;

  // ---- Pass A: iter0 weighted sum; softmax(0) == uniform 1/NNODE, so chain
  //      WMMA accumulation over all n (two chains for ILP), scale at the end.
  //      Constant trip count -> scalar loop, EXEC untouched around WMMA.
  v8f accA = (v8f)0.0f;
  v8f accB = (v8f)0.0f;
  for (int k = 0; k < KITERS; k += 2) {
    v2f a0, a1, bm0, bm1;
    const int n0 = wave + k * NWAVES;
    load_ab(xbase, wbase, n0, a0, a1, bm0, bm1);
    accA = wmma4(a0, bm0, accA);
    accA = wmma4(a1, bm1, accA);
    load_ab(xbase, wbase, n0 + NWAVES, a0, a1, bm0, bm1);
    accB = wmma4(a0, bm0, accB);
    accB = wmma4(a1, bm1, accB);
  }
#pragma unroll
  for (int j = 0; j < 8; ++j) accA[j] += accB[j];
  reduce_squash(accA, 1.0f / (float)NNODE, sred, sArr, outArr, scaleArr, tid,
                lane, wave);

  // ---- Two more routing iterations: agreement update + weighted sum
#pragma unroll
  for (int it = 0; it < 2; ++it) {
    // agreement pass: logits[n][b] (+)= sum_o prior[n][b][o]*output[b][o]
    float outv[8];
#pragma unroll
    for (int j = 0; j < 8; ++j) outv[j] = outArr[(j + 8 * half) * 16 + laneLo];
    for (int k = 0; k < KITERS; ++k) {
      const int n = wave + k * NWAVES;
      v2f a0, a1, bm0, bm1;
      load_ab(xbase, wbase, n, a0, a1, bm0, bm1);
      v8f p = (v8f)0.0f;
      p = wmma4(a0, bm0, p);
      p = wmma4(a1, bm1, p);
#pragma unroll
      for (int j = 0; j < 8; ++j) {
        float t = half16_sum(p[j] * outv[j]);
        if (laneLo == 0) {
          const int idx = n * 16 + j + 8 * half;
          lg[idx] = (it == 0) ? t : (lg[idx] + t);
        }
      }
    }
    __syncthreads();

    softmax_n(lg, probs, red, maxb, invZ, tid);

    // weighted-sum pass: s[b][o] = sum_n probs[n][b]*prior[n][b][o]
    v8f acc = (v8f)0.0f;
    for (int k = 0; k < KITERS; ++k) {
      const int n = wave + k * NWAVES;
      v2f a0, a1, bm0, bm1;
      load_ab(xbase, wbase, n, a0, a1, bm0, bm1);
      v8f p = (v8f)0.0f;
      p = wmma4(a0, bm0, p);
      p = wmma4(a1, bm1, p);
      const float4 pr01 = *(const float4*)(probs + n * 16 + half * 8);
      const float4 pr23 = *(const float4*)(probs + n * 16 + half * 8 + 4);
      acc[0] = fmaf(pr01.x, p[0], acc[0]);
      acc[1] = fmaf(pr01.y, p[1], acc[1]);
      acc[2] = fmaf(pr01.z, p[2], acc[2]);
      acc[3] = fmaf(pr01.w, p[3], acc[3]);
      acc[4] = fmaf(pr23.x, p[4], acc[4]);
      acc[5] = fmaf(pr23.y, p[5], acc[5]);
      acc[6] = fmaf(pr23.z, p[6], acc[6]);
      acc[7] = fmaf(pr23.w, p[7], acc[7]);
    }
    reduce_squash(acc, 1.0f, sred, sArr, outArr, scaleArr, tid, lane, wave);
  }

  // out[c][b0+b][0][o], coalesced: flat = c*4096 + b0*16 + tid
  out[(size_t)c * (NBATCH * DOUT) + (size_t)b0 * DOUT + tid] = outArr[tid];
}

extern "C" void kernel_launch(void* const* d_in, const int* in_sizes, int n_in,
                              void* d_out, int out_size, void* d_ws,
                              size_t ws_size, hipStream_t stream) {
  (void)in_sizes; (void)n_in; (void)d_ws; (void)ws_size; (void)out_size;
  const float* x = (const float*)d_in[0];
  const float* W = (const float*)d_in[1];
  float* out = (float*)d_out;
  dim3 grid(NCAP * (NBATCH / BTILE));  // 160 workgroups
  dim3 block(256);                     // 8 wave32s
  hipLaunchKernelGGL(capsule_route_kernel, grid, block, 0, stream, x, W, out);
}